// Gaussian_Distance_3453153706461
// MI455X (gfx1250) — compile-verified
//
#include <hip/hip_runtime.h>
#include <math.h>

#define N_PTS   1024
#define DIM     128
#define TWO_N   2048
#define BT      64      // tile rows/cols
#define DCH     32      // d-chunk width staged in LDS
#define PITCH   36      // LDS row pitch in floats (bank-conflict-free with strided row map)
#define TILES   32      // TWO_N / BT
#define NBLK    528     // TILES*(TILES+1)/2 upper-triangular tiles

typedef int v4i __attribute__((__vector_size__(16)));
typedef __attribute__((address_space(1))) v4i* gv4i_ptr;
typedef __attribute__((address_space(3))) v4i* lv4i_ptr;

// ---------- fast math primitives (single-instruction on gfx1250) ----------
__device__ __forceinline__ float fast_rsqrt(float x) {
#if __has_builtin(__builtin_amdgcn_rsqf)
    return __builtin_amdgcn_rsqf(x);
#else
    return rsqrtf(x);
#endif
}
__device__ __forceinline__ float fast_exp2(float x) {
#if __has_builtin(__builtin_amdgcn_exp2f)
    return __builtin_amdgcn_exp2f(x);
#else
    return exp2f(x);
#endif
}

// ---------- CDNA5 async global->LDS staging (16B per lane) ----------
__device__ __forceinline__ void stage16(const float* __restrict__ g, float* l) {
#if __has_builtin(__builtin_amdgcn_global_load_async_to_lds_b128)
    __builtin_amdgcn_global_load_async_to_lds_b128(
        (gv4i_ptr)g,
        (lv4i_ptr)l,
        /*offset=*/0, /*cpol=*/0);
#else
    *(float4*)l = *(const float4*)g;
#endif
}
__device__ __forceinline__ void stage_wait() {
#if __has_builtin(__builtin_amdgcn_global_load_async_to_lds_b128)
#if __has_builtin(__builtin_amdgcn_s_wait_asynccnt)
    __builtin_amdgcn_s_wait_asynccnt(0);
#else
    asm volatile("s_wait_asynccnt 0" ::: "memory");
#endif
#endif
}

// ---------- kernel 1: concatenate + var = exp(logvar) ----------
__global__ __launch_bounds__(256) void pack_kernel(
    const float* __restrict__ mu_a, const float* __restrict__ lva,
    const float* __restrict__ mu_b, const float* __restrict__ lvb,
    float* __restrict__ mu_cat, float* __restrict__ var_cat) {
    const int idx  = blockIdx.x * 256 + threadIdx.x;   // grid covers exactly 2048*128
    const int half = N_PTS * DIM;
    float mu, lv;
    if (idx < half) { mu = mu_a[idx];        lv = lva[idx];        }
    else            { mu = mu_b[idx - half]; lv = lvb[idx - half]; }
    mu_cat[idx]  = mu;
    var_cat[idx] = fast_exp2(lv * 1.44269504f);        // exp(lv) = exp2(lv*log2e)
}

// ---------- kernel 2: upper-triangular 64x64 pair tiles ----------
__global__ __launch_bounds__(256) void pair_kernel(
    const float* __restrict__ mu_cat, const float* __restrict__ var_cat,
    float* __restrict__ partials) {
    __shared__ float sMuR[BT][PITCH];
    __shared__ float sVaR[BT][PITCH];
    __shared__ float sMuC[BT][PITCH];
    __shared__ float sVaC[BT][PITCH];
    __shared__ float red[256];

    // map linear block id -> upper-triangular tile (bi <= bj)
    int b = blockIdx.x;
    int bi = 0, rowlen = TILES, rem = b;
    while (rem >= rowlen) { rem -= rowlen; rowlen--; bi++; }
    const int bj = bi + rem;

    const int tid = threadIdx.x;
    const int tx  = tid & 15;    // column group
    const int ty  = tid >> 4;    // row group
    const int gRow0 = bi * BT;
    const int gCol0 = bj * BT;

    const float C = 0.84932181f; // sqrt(0.5 * log2(e)) -> folds -0.5 and log2e into one mul
    float acc = 0.0f;

    for (int ch = 0; ch < DIM / DCH; ++ch) {
        const int d0 = ch * DCH;
        // cooperative async staging: 4 arrays x 64 rows x 32 floats, 16B per lane-op
#pragma unroll
        for (int t = 0; t < 2; ++t) {
            const int idx = tid + 256 * t;     // 512 groups of 4 floats per array
            const int row = idx >> 3;
            const int q4  = (idx & 7) * 4;
            stage16(&mu_cat [(gRow0 + row) * DIM + d0 + q4], &sMuR[row][q4]);
            stage16(&var_cat[(gRow0 + row) * DIM + d0 + q4], &sVaR[row][q4]);
            stage16(&mu_cat [(gCol0 + row) * DIM + d0 + q4], &sMuC[row][q4]);
            stage16(&var_cat[(gCol0 + row) * DIM + d0 + q4], &sVaC[row][q4]);
        }
        stage_wait();
        __syncthreads();

#pragma unroll 2
        for (int dd = 0; dd < DCH; ++dd) {
            float m1[4], w1[4], m2[4], w2[4];
#pragma unroll
            for (int c = 0; c < 4; ++c) {
                m1[c] = sMuR[ty + 16 * c][dd];   // broadcast across tx
                w1[c] = sVaR[ty + 16 * c][dd];
                m2[c] = sMuC[tx + 16 * c][dd];   // 16 distinct banks (pitch 36, stride-16 rows)
                w2[c] = sVaC[tx + 16 * c][dd];
            }
#pragma unroll
            for (int ci = 0; ci < 4; ++ci) {
#pragma unroll
                for (int cj = 0; cj < 4; ++cj) {
                    const float t  = m1[ci] - m2[cj];
                    const float vs = w1[ci] + w2[cj];
                    const float r  = fast_rsqrt(vs);      // v_rsq_f32
                    const float th = t * C;
                    const float a  = th * th;             // 0.5*log2e*t^2
                    const float r2 = r * r;               // 1/vs
                    const float e  = fast_exp2(-(a * r2)); // v_exp_f32
                    acc = fmaf(e, r, acc);
                }
            }
        }
        __syncthreads();   // all waves done reading before next chunk's async writes land
    }

    // deterministic block reduction
    red[tid] = acc;
    __syncthreads();
    for (int s = 128; s > 0; s >>= 1) {
        if (tid < s) red[tid] += red[tid + s];
        __syncthreads();
    }
    if (tid == 0) {
        const float sgn  = ((gRow0 < N_PTS) == (gCol0 < N_PTS)) ? 1.0f : -1.0f;
        const float mult = (bi == bj) ? 1.0f : 2.0f;
        partials[b] = red[0] * sgn * mult * 7.4505805969238281e-9f; // exact 2^-27
    }
}

// ---------- kernel 3: deterministic final reduction ----------
__global__ __launch_bounds__(256) void final_reduce(
    const float* __restrict__ partials, float* __restrict__ out, int n) {
    __shared__ float red[256];
    float s = 0.0f;
    for (int i = threadIdx.x; i < n; i += 256) s += partials[i];
    red[threadIdx.x] = s;
    __syncthreads();
    for (int st = 128; st > 0; st >>= 1) {
        if (threadIdx.x < st) red[threadIdx.x] += red[threadIdx.x + st];
        __syncthreads();
    }
    if (threadIdx.x == 0) out[0] = red[0];
}

extern "C" void kernel_launch(void* const* d_in, const int* in_sizes, int n_in,
                              void* d_out, int out_size, void* d_ws, size_t ws_size,
                              hipStream_t stream) {
    (void)in_sizes; (void)n_in; (void)out_size; (void)ws_size;
    const float* mu_a = (const float*)d_in[0];
    const float* lva  = (const float*)d_in[1];
    const float* mu_b = (const float*)d_in[2];
    const float* lvb  = (const float*)d_in[3];

    float* ws       = (float*)d_ws;
    float* mu_cat   = ws;                       // 2048*128 floats = 1 MB
    float* var_cat  = ws + TWO_N * DIM;         // 1 MB
    float* partials = ws + 2 * TWO_N * DIM;     // 528 floats

    pack_kernel<<<(TWO_N * DIM) / 256, 256, 0, stream>>>(mu_a, lva, mu_b, lvb,
                                                         mu_cat, var_cat);
    pair_kernel<<<NBLK, 256, 0, stream>>>(mu_cat, var_cat, partials);
    final_reduce<<<1, 256, 0, stream>>>(partials, (float*)d_out, NBLK);
}